// TOLD_41824391528924
// MI455X (gfx1250) — compile-verified
//
#include <hip/hip_runtime.h>
#include <math.h>

#define HOR     18
#define NSAMP   65536
#define ADIM    4
#define NELITE  64
#define NITER   6
#define TOT     (2 * NSAMP)

typedef __attribute__((ext_vector_type(2))) float v2f;
typedef __attribute__((ext_vector_type(8))) float v8f;

// ---------------------------------------------------------------------------
// mean <- pi_actions[:, 0, :]
// ---------------------------------------------------------------------------
__global__ __launch_bounds__(128) void mppi_init_mean(const float* __restrict__ pi,
                                                      float* __restrict__ mean) {
  int t = threadIdx.x;
  if (t < HOR * ADIM) {
    int h = t >> 2, a = t & 3;
    mean[t] = pi[(size_t)h * NSAMP * ADIM + a];   // sample index 0
  }
}

// ---------------------------------------------------------------------------
// value[n] = -sum_{h,a} (x[h,n,a] - t[h,a])^2  =  2*(x.t) - |x|^2 - |t|^2
// x.t via V_WMMA_F32_16X16X4_F32, two 16-sample tiles per wave (32 samples),
// K=4 per horizon step, 2x18 chained WMMAs on f32 accumulators.  CEM is a
// template parameter so both variants are branch-free straight-line code
// (no per-step s_wait_loadcnt 0 drains, loads fully pipelined).
// ---------------------------------------------------------------------------
template <int CEM>
__global__ __launch_bounds__(256) void mppi_value(const float* __restrict__ src,
                                                  const float* __restrict__ mean,
                                                  const float* __restrict__ target,
                                                  float* __restrict__ vout,
                                                  float scale) {
  __shared__ float dsh[8][32];                 // per-wave D staging (2 tiles)
  const int tid  = threadIdx.x;
  const int wave = tid >> 5;
  const int lane = tid & 31;
  const int m    = lane & 15;                  // row within tile
  const int half = lane >> 4;                  // which pair of action comps
  const int base = (blockIdx.x * 8 + wave) * 32;
  const int s0   = base + m;                   // tile 0 sample
  const int s1   = base + 16 + m;              // tile 1 sample

  v8f c0 = {0.f, 0.f, 0.f, 0.f, 0.f, 0.f, 0.f, 0.f};
  v8f c1 = {0.f, 0.f, 0.f, 0.f, 0.f, 0.f, 0.f, 0.f};
  float p0 = 0.f, p1 = 0.f;                    // partial |x|^2 (2 of 4 comps)
  float q  = 0.f;                              // partial |t|^2
#pragma unroll
  for (int h = 0; h < HOR; ++h) {
    v2f x0 = *(const v2f*)(src + ((size_t)h * NSAMP + s0) * ADIM + 2 * half);
    v2f x1 = *(const v2f*)(src + ((size_t)h * NSAMP + s1) * ADIM + 2 * half);
    v2f tt = *(const v2f*)(target + h * ADIM + 2 * half);
    if (CEM) {
      v2f mu = *(const v2f*)(mean + h * ADIM + 2 * half);
      x0.x = fminf(fmaxf(fmaf(scale, x0.x, mu.x), -4.f), 4.f);
      x0.y = fminf(fmaxf(fmaf(scale, x0.y, mu.y), -4.f), 4.f);
      x1.x = fminf(fmaxf(fmaf(scale, x1.x, mu.x), -4.f), 4.f);
      x1.y = fminf(fmaxf(fmaf(scale, x1.y, mu.y), -4.f), 4.f);
    }
    p0 = fmaf(x0.x, x0.x, p0);  p0 = fmaf(x0.y, x0.y, p0);
    p1 = fmaf(x1.x, x1.x, p1);  p1 = fmaf(x1.y, x1.y, p1);
    q  = fmaf(tt.x, tt.x, q);   q  = fmaf(tt.y, tt.y, q);
    // D += A(16x4 f32) x B(4x16 f32, columns = broadcast target)
    c0 = __builtin_amdgcn_wmma_f32_16x16x4_f32(false, x0, false, tt,
                                               (short)0, c0, false, false);
    c1 = __builtin_amdgcn_wmma_f32_16x16x4_f32(false, x1, false, tt,
                                               (short)0, c1, false, false);
  }
  // C layout: VGPR r = row M=r (lanes 0-15) / M=8+r (lanes 16-31).
  // Column 0 suffices (all columns identical): lanes 0 and 16 export D.
  if (m == 0) {
#pragma unroll
    for (int r = 0; r < 8; ++r) {
      dsh[wave][half * 8 + r]      = c0[r];
      dsh[wave][16 + half * 8 + r] = c1[r];
    }
  }
  __syncthreads();
  float S20 = p0 + __shfl_xor(p0, 16, 32);     // full |x|^2 per sample
  float S21 = p1 + __shfl_xor(p1, 16, 32);
  float T2  = q  + __shfl_xor(q, 16, 32);      // full |t|^2
  if (half == 0) {
    vout[s0] = 2.f * dsh[wave][m]      - S20 - T2;
    vout[s1] = 2.f * dsh[wave][16 + m] - S21 - T2;
  }
}

// ---------------------------------------------------------------------------
// Single-block top-64 + softmax weights + elite gather + mean update.
// Each thread owns 128 strided values; extraction removes only from the
// owner's private chunk via a 128-bit register mask (no global coherence
// games).  Tie-break: lowest global index, matching jax.lax.top_k.
// ---------------------------------------------------------------------------
__global__ __launch_bounds__(1024) void mppi_topk(const float* __restrict__ value,
                                                  const float* __restrict__ pi,
                                                  const float* __restrict__ noise_i,
                                                  float* __restrict__ mean,
                                                  float scale, int last,
                                                  float* __restrict__ out) {
  __shared__ float sVal[1024];  __shared__ int sIdx[1024];
  __shared__ float rVal[1024];  __shared__ int rIdx[1024];
  __shared__ float eV[NELITE];  __shared__ int eI[NELITE];
  __shared__ float wgt[NELITE];
  __shared__ float sDen;
  const int t = threadIdx.x;

  uint32_t ex0 = 0, ex1 = 0, ex2 = 0, ex3 = 0;   // 128-slot exclusion mask
  float lm = -INFINITY; int lmi = 0;
  for (int s = 0; s < TOT / 1024; ++s) {          // strict > keeps lowest idx
    float v = value[t + s * 1024];
    if (v > lm) { lm = v; lmi = t + s * 1024; }
  }
  sVal[t] = lm; sIdx[t] = lmi;
  __syncthreads();

  for (int r = 0; r < NELITE; ++r) {
    rVal[t] = sVal[t]; rIdx[t] = sIdx[t];
    __syncthreads();
    for (int stride = 512; stride > 0; stride >>= 1) {
      if (t < stride) {
        float v2 = rVal[t + stride]; int i2 = rIdx[t + stride];
        if (v2 > rVal[t] || (v2 == rVal[t] && i2 < rIdx[t])) {
          rVal[t] = v2; rIdx[t] = i2;
        }
      }
      __syncthreads();
    }
    const int widx = rIdx[0];
    if (t == 0) { eV[r] = rVal[0]; eI[r] = widx; }
    if (t == (widx & 1023)) {                     // only the owner rescans
      int slot = widx >> 10;
      if      (slot < 32) ex0 |= 1u << slot;
      else if (slot < 64) ex1 |= 1u << (slot - 32);
      else if (slot < 96) ex2 |= 1u << (slot - 64);
      else                ex3 |= 1u << (slot - 96);
      float nm = -INFINITY; int nmi = 0;
      for (int s = 0; s < TOT / 1024; ++s) {
        uint32_t bit = (s < 32) ? (ex0 >> s)
                     : (s < 64) ? (ex1 >> (s - 32))
                     : (s < 96) ? (ex2 >> (s - 64))
                                : (ex3 >> (s - 96));
        if (bit & 1u) continue;
        float v = value[t + s * 1024];
        if (v > nm) { nm = v; nmi = t + s * 1024; }
      }
      sVal[t] = nm; sIdx[t] = nmi;
    }
    __syncthreads();
  }

  // softmax weights (temperature 1, max = first extracted value)
  if (t < NELITE) wgt[t] = expf(eV[t] - eV[0]);
  __syncthreads();
  if (t == 0) {
    float s1 = 0.f;
    for (int e = 0; e < NELITE; ++e) s1 += wgt[e];
    float s2 = 0.f;
    for (int e = 0; e < NELITE; ++e) { wgt[e] /= s1; s2 += wgt[e]; }
    sDen = s2 + 1e-9f;                            // mirrors (ssum + 1e-9)
  }
  __syncthreads();

  // weighted elite mean -> momentum mean update; last iter: emit argmax traj
  if (t < HOR * ADIM) {
    const int h = t >> 2, a = t & 3;
    const float mold = mean[t];
    float acc = 0.f, top = 0.f;
    for (int e = 0; e < NELITE; ++e) {
      const int idx = eI[e];
      float x;
      if (idx < NSAMP) {
        x = pi[((size_t)h * NSAMP + idx) * ADIM + a];
      } else {
        float nz = noise_i[((size_t)h * NSAMP + (idx - NSAMP)) * ADIM + a];
        x = fminf(fmaxf(fmaf(scale, nz, mold), -4.f), 4.f);
      }
      if (e == 0) top = x;
      acc = fmaf(wgt[e], x, acc);
    }
    mean[t] = 0.1f * mold + 0.9f * (acc / sDen);
    if (last) out[t] = top;
  }
}

// ---------------------------------------------------------------------------
extern "C" void kernel_launch(void* const* d_in, const int* in_sizes, int n_in,
                              void* d_out, int out_size, void* d_ws, size_t ws_size,
                              hipStream_t stream) {
  const float* pi     = (const float*)d_in[0];   // (18, 65536, 4)
  const float* noise  = (const float*)d_in[1];   // (6, 18, 65536, 4)
  const float* target = (const float*)d_in[2];   // (18, 4)
  float* out  = (float*)d_out;                   // (18, 4)
  float* val  = (float*)d_ws;                    // TOT floats
  float* mean = val + TOT;                       // 72 floats

  mppi_init_mean<<<1, 128, 0, stream>>>(pi, mean);
  // pi-half values are iteration-invariant: compute once.
  mppi_value<0><<<NSAMP / 256, 256, 0, stream>>>(pi, target /*unused*/, target,
                                                 val, 0.f);
  for (int i = 0; i < NITER; ++i) {
    const float* ni = noise + (size_t)i * HOR * NSAMP * ADIM;
    // std: ones at iter 0, clip(_std,18,2)==2.0 afterwards -> 0.5*std scale:
    float scale = (i == 0) ? 0.5f : 1.0f;
    mppi_value<1><<<NSAMP / 256, 256, 0, stream>>>(ni, mean, target,
                                                   val + NSAMP, scale);
    mppi_topk<<<1, 1024, 0, stream>>>(val, pi, ni, mean, scale,
                                      (i == NITER - 1) ? 1 : 0, out);
  }
}